// GATv2_79963701117623
// MI455X (gfx1250) — compile-verified
//
#include <hip/hip_runtime.h>
#include <hip/hip_bf16.h>

#define NHEAD  8
#define CHID   8
#define ALPHA  0.2f
#define B_     2
#define T_     12
#define N_     207
#define CIN    64
#define HC     64          // NHEAD*CHID
#define BT     (B_*T_)     // 24
#define MROWS  (BT*N_)     // 4968
#define MTILES ((MROWS + 15)/16)   // 311

typedef float v2f __attribute__((ext_vector_type(2)));
typedef float v8f __attribute__((ext_vector_type(8)));

static __device__ inline v8f zero8() {
    v8f z;
    #pragma unroll
    for (int v = 0; v < 8; ++v) z[v] = 0.0f;
    return z;
}

// ---------------------------------------------------------------------------
// Kernel 1: x_l = xt@Wl + bl,  x_r = xt@Wr + br   via V_WMMA_F32_16X16X4_F32
// xt[r, k] = x[b, k, n, t]  with r = (b*T+t)*N + n  (the reference transpose)
// One wave per 16-row M tile; 8 16x16 output tiles (4 l + 4 r) accumulated in
// registers over K=64 in steps of 4 (128 wmma issues per wave).
// ---------------------------------------------------------------------------
__global__ __launch_bounds__(32) void gat_proj_wmma(
    const float* __restrict__ x,    // [B, CIN, N, T]
    const float* __restrict__ Wl,   // [CIN, HC]
    const float* __restrict__ bl,   // [HC]
    const float* __restrict__ Wr,   // [CIN, HC]
    const float* __restrict__ br,   // [HC]
    float* __restrict__ xl,         // [MROWS, HC]
    float* __restrict__ xr)         // [MROWS, HC]
{
    __shared__ float Atile[16][CIN + 4];

    const int tile = blockIdx.x;
    const int lane = threadIdx.x;            // 0..31 (wave32)

    // Gather the 16xCIN A-tile into LDS (handles the [B,C,N,T] transpose).
    #pragma unroll
    for (int it = 0; it < (16 * CIN) / 32; ++it) {
        int idx = it * 32 + lane;
        int rr  = idx >> 6;                  // row in tile
        int k   = idx & 63;                  // K index
        int r   = tile * 16 + rr;
        float v = 0.0f;
        if (r < MROWS) {
            int bt = r / N_;
            int n  = r - bt * N_;
            int b  = bt / T_;
            int t  = bt - b * T_;
            v = x[((b * CIN + k) * N_ + n) * T_ + t];
        }
        Atile[rr][k] = v;
    }
    __syncthreads();

    v8f accL[4], accR[4];
    #pragma unroll
    for (int nt = 0; nt < 4; ++nt) { accL[nt] = zero8(); accR[nt] = zero8(); }

    const int halff = lane >> 4;             // 0 or 1
    const int l16   = lane & 15;

    for (int k0 = 0; k0 < CIN; k0 += 4) {
        const int ka = k0 + 2 * halff;       // this lane's K for .x (.y = ka+1)
        v2f a;
        a.x = Atile[l16][ka];
        a.y = Atile[l16][ka + 1];
        #pragma unroll
        for (int nt = 0; nt < 4; ++nt) {
            const int col = nt * 16 + l16;
            v2f bL, bR;
            bL.x = Wl[ka * HC + col];
            bL.y = Wl[(ka + 1) * HC + col];
            bR.x = Wr[ka * HC + col];
            bR.y = Wr[(ka + 1) * HC + col];
            accL[nt] = __builtin_amdgcn_wmma_f32_16x16x4_f32(
                false, a, false, bL, (short)0, accL[nt], false, false);
            accR[nt] = __builtin_amdgcn_wmma_f32_16x16x4_f32(
                false, a, false, bR, (short)0, accR[nt], false, false);
        }
    }

    // D layout: VGPR v -> row v + 8*(lane/16), col = lane%16 (per 16-col tile)
    const int rbase = tile * 16 + 8 * halff;
    if ((tile + 1) * 16 <= MROWS) {
        // fast path: full tile, no guards
        #pragma unroll
        for (int nt = 0; nt < 4; ++nt) {
            const int col = nt * 16 + l16;
            const float biasL = bl[col];
            const float biasR = br[col];
            #pragma unroll
            for (int v = 0; v < 8; ++v) {
                const int r = rbase + v;
                xl[r * HC + col] = accL[nt][v] + biasL;
                xr[r * HC + col] = accR[nt][v] + biasR;
            }
        }
    } else {
        // tail tile (only the last of 311)
        #pragma unroll
        for (int nt = 0; nt < 4; ++nt) {
            const int col = nt * 16 + l16;
            const float biasL = bl[col];
            const float biasR = br[col];
            #pragma unroll
            for (int v = 0; v < 8; ++v) {
                const int r = rbase + v;
                if (r < MROWS) {
                    xl[r * HC + col] = accL[nt][v] + biasL;
                    xr[r * HC + col] = accR[nt][v] + biasR;
                }
            }
        }
    }
}

// ---------------------------------------------------------------------------
// Kernel 2: scores + mask + softmax over j + aggregation + output transpose.
// One wave per (bt, i); lanes parallel over neighbors j; e cached in LDS
// (b128 traffic); LDS transpose reduction for the 64 output accumulators so
// each lane owns 2 outputs and stores run in parallel.
// ---------------------------------------------------------------------------
__global__ __launch_bounds__(32) void gat_attn(
    const float* __restrict__ xl,      // [MROWS, HC]
    const float* __restrict__ xr,      // [MROWS, HC]
    const int*   __restrict__ adj,     // [N, N]
    const float* __restrict__ attn_w,  // [CHID]
    const float* __restrict__ attn_b,  // [1]
    float* __restrict__ out)           // [B, HC, N, T]
{
    __shared__ float eSh[N_ * NHEAD];
    __shared__ float red[32][HC + 4];     // +4 pad: conflict-free transpose

    const int wid  = blockIdx.x;          // 0 .. BT*N-1
    const int bt   = wid / N_;
    const int i    = wid - bt * N_;
    const int lane = threadIdx.x;

    // x_r[bt, i, :] -> registers (same address for all lanes: L0 broadcast)
    float xri[HC];
    {
        const float4* __restrict__ p4 = (const float4*)&xr[(bt * N_ + i) * HC];
        #pragma unroll
        for (int q = 0; q < HC / 4; ++q) {
            float4 v = p4[q];
            xri[4*q]   = v.x; xri[4*q+1] = v.y;
            xri[4*q+2] = v.z; xri[4*q+3] = v.w;
        }
    }

    float aw[CHID];
    #pragma unroll
    for (int c = 0; c < CHID; ++c) aw[c] = attn_w[c];
    const float ab = attn_b[0];
    const float NEG_INF = -__builtin_inff();

    // ---- pass 1: scores + mask + per-head running max ----
    float mx[NHEAD];
    #pragma unroll
    for (int h = 0; h < NHEAD; ++h) mx[h] = NEG_INF;

    for (int jb = 0; jb < N_; jb += 32) {
        const int j = jb + lane;
        if (j < N_) {
            const float4* __restrict__ xlj4 = (const float4*)&xl[(bt * N_ + j) * HC];
            float eh[NHEAD];
            #pragma unroll
            for (int h = 0; h < NHEAD; ++h) {
                float4 p0 = xlj4[2*h];
                float4 p1 = xlj4[2*h + 1];
                float sv[CHID] = {p0.x, p0.y, p0.z, p0.w, p1.x, p1.y, p1.z, p1.w};
                float e = ab;
                #pragma unroll
                for (int c = 0; c < CHID; ++c) {
                    float s = xri[h * CHID + c] + sv[c];
                    s = (s >= 0.0f) ? s : ALPHA * s;
                    e = fmaf(aw[c], s, e);
                }
                // mask[i,j,h] = adj[i, (j*H + h) mod N]
                const int mcol = (j * NHEAD + h) % N_;
                if (adj[i * N_ + mcol] == 0) e = NEG_INF;
                eh[h] = e;
                mx[h] = fmaxf(mx[h], e);
            }
            float4* __restrict__ eDst = (float4*)&eSh[j * NHEAD];
            float4 w0; w0.x = eh[0]; w0.y = eh[1]; w0.z = eh[2]; w0.w = eh[3];
            float4 w1; w1.x = eh[4]; w1.y = eh[5]; w1.z = eh[6]; w1.w = eh[7];
            eDst[0] = w0;
            eDst[1] = w1;
        }
    }
    #pragma unroll
    for (int h = 0; h < NHEAD; ++h)
        for (int off = 16; off > 0; off >>= 1)
            mx[h] = fmaxf(mx[h], __shfl_xor(mx[h], off, 32));

    // ---- pass 2: exp, normalizer, and aggregation of x_r over j ----
    float ssum[NHEAD];
    float acc[HC];
    #pragma unroll
    for (int h = 0; h < NHEAD; ++h) ssum[h] = 0.0f;
    #pragma unroll
    for (int o = 0; o < HC; ++o) acc[o] = 0.0f;

    for (int jb = 0; jb < N_; jb += 32) {
        const int j = jb + lane;
        if (j < N_) {
            const float4* __restrict__ xrj4 = (const float4*)&xr[(bt * N_ + j) * HC];
            const float4* __restrict__ eSrc = (const float4*)&eSh[j * NHEAD];
            float4 e0 = eSrc[0];
            float4 e1 = eSrc[1];
            float ev[NHEAD] = {e0.x, e0.y, e0.z, e0.w, e1.x, e1.y, e1.z, e1.w};
            #pragma unroll
            for (int h = 0; h < NHEAD; ++h) {
                const float p = __expf(ev[h] - mx[h]);
                ssum[h] += p;
                float4 r0 = xrj4[2*h];
                float4 r1 = xrj4[2*h + 1];
                acc[h*CHID + 0] = fmaf(p, r0.x, acc[h*CHID + 0]);
                acc[h*CHID + 1] = fmaf(p, r0.y, acc[h*CHID + 1]);
                acc[h*CHID + 2] = fmaf(p, r0.z, acc[h*CHID + 2]);
                acc[h*CHID + 3] = fmaf(p, r0.w, acc[h*CHID + 3]);
                acc[h*CHID + 4] = fmaf(p, r1.x, acc[h*CHID + 4]);
                acc[h*CHID + 5] = fmaf(p, r1.y, acc[h*CHID + 5]);
                acc[h*CHID + 6] = fmaf(p, r1.z, acc[h*CHID + 6]);
                acc[h*CHID + 7] = fmaf(p, r1.w, acc[h*CHID + 7]);
            }
        }
    }
    // per-head normalizer: wave32 butterfly (8 values, 40 shuffles)
    #pragma unroll
    for (int h = 0; h < NHEAD; ++h)
        for (int off = 16; off > 0; off >>= 1)
            ssum[h] += __shfl_xor(ssum[h], off, 32);

    // 64-accumulator reduction: LDS transpose (padded rows -> conflict-free)
    #pragma unroll
    for (int o4 = 0; o4 < HC; o4 += 4) {
        float4 w; w.x = acc[o4]; w.y = acc[o4+1]; w.z = acc[o4+2]; w.w = acc[o4+3];
        *(float4*)&red[lane][o4] = w;
    }
    __syncthreads();
    float r0 = 0.0f, r1 = 0.0f;
    #pragma unroll
    for (int l = 0; l < 32; ++l) {
        r0 += red[l][lane];
        r1 += red[l][lane + 32];
    }

    // select this lane's per-head denominators without dynamic reg indexing
    const int h0 = lane >> 3;             // head of output o0 = lane
    const int h1 = (lane + 32) >> 3;      // head of output o1 = lane + 32
    float den0 = ssum[0], den1 = ssum[0];
    #pragma unroll
    for (int h = 1; h < NHEAD; ++h) {
        den0 = (h0 == h) ? ssum[h] : den0;
        den1 = (h1 == h) ? ssum[h] : den1;
    }

    // out[b, o, n=i, t] with bt = b*T + t  (reference's final transpose)
    const int b = bt / T_;
    const int t = bt - b * T_;
    out[((b * HC + lane)        * N_ + i) * T_ + t] = r0 / den0;
    out[((b * HC + (lane + 32)) * N_ + i) * T_ + t] = r1 / den1;
}

extern "C" void kernel_launch(void* const* d_in, const int* in_sizes, int n_in,
                              void* d_out, int out_size, void* d_ws, size_t ws_size,
                              hipStream_t stream) {
    const float* x      = (const float*)d_in[0];
    const int*   adj    = (const int*)  d_in[1];
    const float* Wl     = (const float*)d_in[2];
    const float* bl     = (const float*)d_in[3];
    const float* Wr     = (const float*)d_in[4];
    const float* br     = (const float*)d_in[5];
    const float* attn_w = (const float*)d_in[6];
    const float* attn_b = (const float*)d_in[7];

    float* xl = (float*)d_ws;                     // [MROWS, HC]
    float* xr = xl + (size_t)MROWS * HC;          // [MROWS, HC]

    gat_proj_wmma<<<MTILES, 32, 0, stream>>>(x, Wl, bl, Wr, br, xl, xr);
    gat_attn<<<BT * N_, 32, 0, stream>>>(xl, xr, adj, attn_w, attn_b, (float*)d_out);
}